// GIN_82240033784020
// MI455X (gfx1250) — compile-verified
//
#include <hip/hip_runtime.h>
#include <hip/hip_bf16.h>

#define NNODES 50000
#define NEDGES 800000
#define DIM    128
#define NLAYER 3
#define BN_EPS 1e-5f
#define LDS_STRIDE 132   // 128 + 4 pad: strided K-column reads hit distinct banks

typedef __attribute__((ext_vector_type(2))) float v2f;
typedef __attribute__((ext_vector_type(8))) float v8f;

// ---------------------------------------------------------------- zero fill
__global__ void gin_zero_f32(float* __restrict__ p, long n4) {
    long i = (long)blockIdx.x * blockDim.x + threadIdx.x;
    if (i < n4) ((float4*)p)[i] = make_float4(0.f, 0.f, 0.f, 0.f);
}

// ----------------------------------------------------- 128x128 W transpose
// WT[c][k] = W[k][c]; makes each lane's B fragment (2 consecutive K) a b64.
__global__ void gin_transpose128(const float* __restrict__ W,
                                 float* __restrict__ WT) {
    int c = blockIdx.x;      // 0..127 column of W
    int k = threadIdx.x;     // 0..127 row of W
    WT[(size_t)c * DIM + k] = W[(size_t)k * DIM + c];
}

// ------------------------------------------------------- edge scatter (agg)
// one wave32 per edge; each lane moves 4 consecutive floats (32*4 = 128 = D)
__global__ void gin_scatter(const float* __restrict__ x,
                            const long long* __restrict__ ei,
                            float* __restrict__ agg, int E) {
    long t = (long)blockIdx.x * blockDim.x + threadIdx.x;
    int e = (int)(t >> 5);
    int l = (int)(t & 31);
    if (e >= E) return;
    long long s = ei[e];                 // src row
    long long d = ei[(long long)E + e];  // dst row
    float4 v = *(const float4*)(x + s * DIM + l * 4);
    float* dp = agg + d * DIM + l * 4;
    atomicAdd(dp + 0, v.x);
    atomicAdd(dp + 1, v.y);
    atomicAdd(dp + 2, v.z);
    atomicAdd(dp + 3, v.w);
}

// ---------------------------------------------------- WMMA fp32 GEMM panel
// One block = 16 output rows x 128 cols. 8 waves, each wave owns a 16x16 tile.
// B comes from the pre-transposed weight WmT (col-major), so each lane's
// fragment is a single contiguous b64 load.
// mode 0: A = (1+eps[layer])*A0 + A1   (GIN combine, A1 = agg)
// mode 1: A = relu(A0*scale[c] + shift[c])  (BN+ReLU fused into A load)
// do_stats: accumulate per-column sum / sum^2 of the output (for BatchNorm).
__global__ __launch_bounds__(256)
void gin_gemm_wmma(const float* __restrict__ A0, const float* __restrict__ A1,
                   const float* __restrict__ WmT, const float* __restrict__ bias,
                   const float* __restrict__ scale, const float* __restrict__ shift,
                   const float* __restrict__ epsArr, int layer,
                   float* __restrict__ Cout,
                   float* __restrict__ ssum, float* __restrict__ ssq,
                   int mode, int do_stats) {
    __shared__ float As[16 * LDS_STRIDE];

    const int tid = threadIdx.x;
    const long rowBase = (long)blockIdx.x * 16;
    const float pre0 = 1.0f + epsArr[layer];

    // L2-resident weight panel: hint prefetch (global_prefetch_b8)
    __builtin_prefetch(WmT + (size_t)tid * 64, 0, 0);

    // Stage A panel (16 x 128) into LDS with fused elementwise transform.
    for (int idx = tid; idx < 16 * DIM; idx += 256) {
        int r = idx >> 7;
        int c = idx & (DIM - 1);
        long g = (rowBase + r) * DIM + c;
        float a;
        if (mode == 0) {
            a = pre0 * A0[g] + A1[g];
        } else {
            a = fmaxf(0.f, A0[g] * scale[c] + shift[c]);
        }
        As[r * LDS_STRIDE + c] = a;
    }
    __syncthreads();

    const int wave = tid >> 5;          // 0..7 -> 16-col slice
    const int lane = tid & 31;
    const int half = lane >> 4;         // 0: K pair {0,1}, 1: K pair {2,3}
    const int l15  = lane & 15;
    const int colg = wave * 16 + l15;   // this lane's output column

    // C/D layout: lane&15 = N; VGPR j = row j (lanes<16) or j+8 (lanes>=16)
    v8f acc;
    {
        float bv = bias[colg];
#pragma unroll
        for (int j = 0; j < 8; ++j) acc[j] = bv;
    }

    const float* wrow = WmT + (size_t)colg * DIM;   // per-lane B base

    // K loop: 32 steps of V_WMMA_F32_16X16X4_F32
#pragma unroll
    for (int k0 = 0; k0 < DIM; k0 += 4) {
        int k = k0 + 2 * half;
        v2f a, b;
        a.x = As[l15 * LDS_STRIDE + k];
        a.y = As[l15 * LDS_STRIDE + k + 1];
        b = *(const v2f*)(wrow + k);                // one global_load_b64
        acc = __builtin_amdgcn_wmma_f32_16x16x4_f32(
            /*neg_a=*/false, a, /*neg_b=*/false, b,
            /*c_mod=*/(short)0, acc, /*reuse_a=*/false, /*reuse_b=*/false);
    }

    // Epilogue: store + fused BatchNorm statistics
    long rb = rowBase + (half ? 8 : 0);
    float s = 0.f, q = 0.f;
#pragma unroll
    for (int j = 0; j < 8; ++j) {
        float v = acc[j];
        Cout[(rb + j) * DIM + colg] = v;
        s += v;
        q += v * v;
    }
    if (do_stats) {
        atomicAdd(&ssum[colg], s);
        atomicAdd(&ssq[colg], q);
    }
}

// --------------------------------------------- fold BN into scale/shift form
// (h-mean)*rsqrt(var+eps)*g + beta  ==  h*scale + shift
__global__ void gin_bn_finalize(const float* __restrict__ ssum,
                                const float* __restrict__ ssq,
                                const float* __restrict__ g,
                                const float* __restrict__ beta,
                                float* __restrict__ scale,
                                float* __restrict__ shift, float invN) {
    int c = threadIdx.x;
    if (c >= DIM) return;
    float mean = ssum[c] * invN;
    float var  = ssq[c] * invN - mean * mean;
    float sc   = g[c] * rsqrtf(var + BN_EPS);
    scale[c] = sc;
    shift[c] = beta[c] - mean * sc;
}

// ---------------------------------------------- inter-layer BN + ReLU apply
__global__ void gin_bn_relu(const float* __restrict__ in,
                            const float* __restrict__ scale,
                            const float* __restrict__ shift,
                            float* __restrict__ out, long n) {
    long i = (long)blockIdx.x * blockDim.x + threadIdx.x;
    if (i < n) {
        int c = (int)(i & (DIM - 1));
        out[i] = fmaxf(0.f, in[i] * scale[c] + shift[c]);
    }
}

// ---------------------------------------------------------------- launcher
extern "C" void kernel_launch(void* const* d_in, const int* in_sizes, int n_in,
                              void* d_out, int out_size, void* d_ws, size_t ws_size,
                              hipStream_t stream) {
    (void)in_sizes; (void)n_in; (void)out_size; (void)ws_size;

    // inputs per setup_inputs() order
    const float*     x0   = (const float*)d_in[0];
    const long long* ei   = (const long long*)d_in[1];
    const float*     eps  = (const float*)d_in[2];
    const float*     W1   = (const float*)d_in[3];
    const float*     b1   = (const float*)d_in[4];
    const float*     g1   = (const float*)d_in[5];
    const float*     bt1  = (const float*)d_in[6];
    const float*     W2   = (const float*)d_in[7];
    const float*     b2   = (const float*)d_in[8];
    const float*     bng  = (const float*)d_in[9];
    const float*     bnb  = (const float*)d_in[10];
    float* out = (float*)d_out;

    const long ND = (long)NNODES * DIM;          // 6.4M floats
    char* ws = (char*)d_ws;
    float* agg   = (float*)ws;                   // N*D  (also GEMM2 temp out)
    float* hbuf  = (float*)(ws + ND * 4);        // N*D
    float* xbuf  = (float*)(ws + 2 * ND * 4);    // N*D
    float* stats = (float*)(ws + 3 * ND * 4);    // 8 * 128 floats
    float* sum1 = stats;        float* sq1 = stats + 128;
    float* sc1  = stats + 256;  float* sh1 = stats + 384;
    float* sum2 = stats + 512;  float* sq2 = stats + 640;
    float* sc2  = stats + 768;  float* sh2 = stats + 896;
    float* WT1  = stats + 1024;                  // 128*128 transposed weights
    float* WT2  = WT1 + DIM * DIM;

    const int DD = DIM * DIM;
    const float invN = 1.0f / (float)NNODES;

    dim3 blk(256);
    const int zeroBlocksND = (int)((ND / 4 + 255) / 256);
    const int scatBlocks   = (int)(((long)NEDGES * 32 + 255) / 256);
    const int gemmBlocks   = NNODES / 16;        // 3125 row panels
    const int elemBlocks   = (int)((ND + 255) / 256);

    for (int i = 0; i < NLAYER; ++i) {
        const float* xcur = (i == 0) ? x0 : xbuf;

        // clear aggregation buffer + BN statistic accumulators
        gin_zero_f32<<<zeroBlocksND, blk, 0, stream>>>(agg, ND / 4);
        gin_zero_f32<<<1, blk, 0, stream>>>(stats, 256);  // sum1,sq1,...,sq2

        // column-major copies of this layer's weights (B-side b64 fragments)
        gin_transpose128<<<DIM, dim3(DIM), 0, stream>>>(W1 + (size_t)i * DD, WT1);
        gin_transpose128<<<DIM, dim3(DIM), 0, stream>>>(W2 + (size_t)i * DD, WT2);

        // sum_{j in N(i)} x_j  via edge-parallel scatter-add
        gin_scatter<<<scatBlocks, blk, 0, stream>>>(xcur, ei, agg, NEDGES);

        // GEMM1: h = ((1+eps)*x + agg) @ W1 + b1, with BN1 stats epilogue
        gin_gemm_wmma<<<gemmBlocks, blk, 0, stream>>>(
            xcur, agg, WT1, b1 + i * DIM,
            nullptr, nullptr, eps, i,
            hbuf, sum1, sq1, /*mode=*/0, /*do_stats=*/1);

        gin_bn_finalize<<<1, dim3(128), 0, stream>>>(sum1, sq1, g1 + i * DIM,
                                                     bt1 + i * DIM, sc1, sh1, invN);

        // GEMM2: x' = relu(bn(h)) @ W2 + b2  (BN+ReLU fused into A load)
        float* gemm2out = (i == NLAYER - 1) ? out : agg;  // agg is free now
        gin_gemm_wmma<<<gemmBlocks, blk, 0, stream>>>(
            hbuf, nullptr, WT2, b2 + i * DIM,
            sc1, sh1, eps, i,
            gemm2out, sum2, sq2, /*mode=*/1, /*do_stats=*/(i < NLAYER - 1) ? 1 : 0);

        if (i < NLAYER - 1) {
            gin_bn_finalize<<<1, dim3(128), 0, stream>>>(sum2, sq2, bng + i * DIM,
                                                         bnb + i * DIM, sc2, sh2, invN);
            gin_bn_relu<<<elemBlocks, blk, 0, stream>>>(agg, sc2, sh2, xbuf, ND);
        }
    }
}